// PatchExtractor3d_28750511079630
// MI455X (gfx1250) — compile-verified
//
#include <hip/hip_runtime.h>

// PatchExtractor3d: out[b, cin*27 + i*9+j*3+l, d, h, w] = x[b, cin, d+i-1, h+j-1, w+l-1]
// x: (2,3,32,128,128) f32.  Output: (2,81,32,128,128) f32 = 339.7MB -> pure store-BW bound
// (ideal ~14.6us @ 23.3TB/s). Input (12.6MB) stays hot in L2/WGP$.
//
// One wave32 per *extended* source row (ds in [-1,32], hs in [-1,128]):
//   - in-range rows: async-DMA 512B row to LDS (global_load_async_to_lds_b128),
//     read back (1 float4/lane), build the two w-shifts with 2 lane shuffles.
//   - out-of-range rows carry zeros and write exactly the boundary-zero output rows.
//   - each wave stores its up-to-27 output rows with NT-hinted global_store_b128.
// All control values are scalarized via readfirstlane so validity tests are s_cbranch
// and stores are SADDR-form: ~1 VMEM + O(1) SALU per 16B of output.

typedef float v4f __attribute__((ext_vector_type(4)));

#define NBC   6            // B*C
#define DEXT  34           // D + 2
#define HEXT  130          // H + 2
#define NROWS (NBC * DEXT * HEXT)          // 26520 extended source rows
#define WPB   8                            // waves per 256-thread block
#define NBLK  (NROWS / WPB)                // 3315 blocks (exact: 3315*8 == 26520)

__global__ __launch_bounds__(256) void patch3d_rowbcast_kernel(
    const float* __restrict__ x, float* __restrict__ out) {
  __shared__ v4f srow[WPB][32];

  const int lane = threadIdx.x & 31;
  // r is wave-uniform (all lanes of a wave share threadIdx.x>>5): force it scalar so
  // the whole control/address pipeline lives in SGPRs.
  const int r = __builtin_amdgcn_readfirstlane(
      (int)(blockIdx.x * WPB) + (int)(threadIdx.x >> 5));
  const int wv  = r & (WPB - 1);             // scalar wave slot (grid is exact)
  const int hs1 = r % HEXT;                  // hs + 1 in [0,129]
  const int t   = r / HEXT;
  const int ds1 = t % DEXT;                  // ds + 1 in [0,33]
  const int bc  = t / DEXT;                  // b*3 + cin in [0,5]
  const int ds  = ds1 - 1;
  const int hs  = hs1 - 1;
  const bool src_ok = ((unsigned)ds < 32u) & ((unsigned)hs < 128u);  // scalar

  v4f v = (v4f){0.f, 0.f, 0.f, 0.f};
  if (src_ok) {                               // uniform s_cbranch
    const v4f* gsrc = reinterpret_cast<const v4f*>(x)
                    + ((size_t)(bc * 32 + ds) * 128 + hs) * 32 + lane;
    // CDNA5 async DMA: global -> LDS, tracked by per-wave ASYNCcnt (no barrier needed).
    unsigned ldsoff = (unsigned)(uintptr_t)(&srow[wv][lane]);  // low 32b of flat = LDS offset
    asm volatile(
        "global_load_async_to_lds_b128 %0, %1, off\n\t"
        "s_wait_asynccnt 0"
        :
        : "v"(ldsoff), "v"(gsrc)
        : "memory");
    v = srow[wv][lane];
    // Warm the next source row of this (bc,ds) plane (gfx1250 global_prefetch_b8).
    if (hs < 127) __builtin_prefetch(gsrc + 32, 0, 1);
  }

  // Build w-1 / w+1 shifted rows with two wave32 shuffles (full EXEC here).
  float top = __shfl_up(v[3], 1, 32);          // x[4*lane - 1]
  float bot = __shfl_down(v[0], 1, 32);        // x[4*lane + 4]
  if (lane == 0)  top = 0.f;                   // w-1 underflow -> pad zero
  if (lane == 31) bot = 0.f;                   // w+1 overflow  -> pad zero
  const v4f rm1 = (v4f){top, v[0], v[1], v[2]};   // l = 0 : x[w-1]
  const v4f rp1 = (v4f){v[1], v[2], v[3], bot};   // l = 2 : x[w+1]

  v4f* const out4 = reinterpret_cast<v4f*>(out);
  const size_t L = (size_t)32 * 128 * 32;      // float4 stride per cout (=131072, 2MB)

#pragma unroll
  for (int i = 0; i < 3; ++i) {
    const int d = ds1 - i;
    if ((unsigned)d >= 32u) continue;          // uniform scalar branch
#pragma unroll
    for (int j = 0; j < 3; ++j) {
      const int h = hs1 - j;
      if ((unsigned)h >= 128u) continue;       // uniform scalar branch
      // Scalar row base; per-lane part is just +lane -> SADDR-form b128 stores.
      v4f* prow = out4
          + ((((size_t)bc * 27 + (i * 9 + j * 3)) * 32 + d) * 128 + h) * 32;
      // Write-once 340MB stream > L2: NT stores keep the hot input resident in L2.
      __builtin_nontemporal_store(rm1, prow + lane);
      __builtin_nontemporal_store(v,   prow + L + lane);
      __builtin_nontemporal_store(rp1, prow + 2 * L + lane);
    }
  }
}

extern "C" void kernel_launch(void* const* d_in, const int* in_sizes, int n_in,
                              void* d_out, int out_size, void* d_ws, size_t ws_size,
                              hipStream_t stream) {
  (void)in_sizes; (void)n_in; (void)d_ws; (void)ws_size; (void)out_size;
  const float* x = (const float*)d_in[0];
  float* out = (float*)d_out;
  patch3d_rowbcast_kernel<<<dim3(NBLK), dim3(256), 0, stream>>>(x, out);
}